// GMM_7370163880609
// MI455X (gfx1250) — compile-verified
//
#include <hip/hip_runtime.h>
#include <math.h>

typedef __attribute__((ext_vector_type(2))) float v2f;
typedef __attribute__((ext_vector_type(4))) float v4f;
typedef __attribute__((ext_vector_type(8))) float v8f;

#define NPTS   500000
#define KCL    16
#define DDIM   16
#define NTILES (NPTS / 16)      // 31250 exactly
#define LOG2PI 1.8378770664093454836f

// ---------------------------------------------------------------------------
// Prep: per-cluster Cholesky, triangular inverse U = L^{-1},
// nb = -(U*mu)  (pre-negated so the main kernel seeds WMMA C directly),
// c = log(pi) - 0.5*D*log(2pi) - sum(log diag L).  One thread per cluster.
// ---------------------------------------------------------------------------
__global__ void gmm_prep_kernel(const float* __restrict__ pi,
                                const float* __restrict__ mu,
                                const float* __restrict__ Sigma,
                                float* __restrict__ Uo,     // (K,16,16)
                                float* __restrict__ nbo,    // (K,16)  = -U*mu
                                float* __restrict__ co) {   // (K)
    int k = threadIdx.x;
    if (k >= KCL) return;
    const float* S = Sigma + k * DDIM * DDIM;

    float L[DDIM][DDIM];
    // Cholesky (lower)
    for (int j = 0; j < DDIM; ++j) {
        float s = S[j * DDIM + j];
        for (int p = 0; p < j; ++p) s -= L[j][p] * L[j][p];
        float d = sqrtf(s);
        L[j][j] = d;
        float inv = 1.0f / d;
        for (int i = j + 1; i < DDIM; ++i) {
            float t = S[i * DDIM + j];
            for (int p = 0; p < j; ++p) t -= L[i][p] * L[j][p];
            L[i][j] = t * inv;
        }
    }
    // U = L^{-1} (lower triangular)
    float U[DDIM][DDIM];
    for (int j = 0; j < DDIM; ++j) {
        for (int i = 0; i < j; ++i) U[i][j] = 0.0f;
        U[j][j] = 1.0f / L[j][j];
        for (int i = j + 1; i < DDIM; ++i) {
            float s = 0.0f;
            for (int p = j; p < i; ++p) s += L[i][p] * U[p][j];
            U[i][j] = -s / L[i][i];
        }
    }
    float hld = 0.0f;
    for (int j = 0; j < DDIM; ++j) hld += logf(L[j][j]);

    const float* m = mu + k * DDIM;           // mu is (K,1,D)
    for (int i = 0; i < DDIM; ++i) {
        float s = 0.0f;
        for (int p = 0; p <= i; ++p) s += U[i][p] * m[p];
        nbo[k * DDIM + i] = -s;               // pre-negated
        for (int p = 0; p < DDIM; ++p) Uo[k * DDIM * DDIM + i * DDIM + p] = U[i][p];
    }
    co[k] = logf(pi[k]) - 0.5f * (float)DDIM * LOG2PI - hld;
}

// ---------------------------------------------------------------------------
// Main: one wave32 per 16-point tile, grid-stride over tiles.  Per cluster k:
//   Z = U_k * X^T - b_k   via 4x V_WMMA_F32_16X16X4_F32 (C seeded with -b_k)
//   maha(point) = sum over rows of Z^2   (8 regs + cross-half shfl_xor 16)
//   logit_k = c_k - 0.5*maha
// then softmax over the 16 logits, coalesced non-temporal store.
//
// An opaque per-iteration zero offset (asm volatile) keeps the sU/sNB reads
// inside the loop as ds_loads: without it, LICM promotes the whole 17 KB
// table into 400+ VGPRs (round 2) or scratch spills (round 3).
// ---------------------------------------------------------------------------
__global__ void __launch_bounds__(256)
gmm_main_kernel(const float* __restrict__ x,
                const float* __restrict__ Ug,
                const float* __restrict__ nbg,
                const float* __restrict__ cg,
                float* __restrict__ out,
                int ntiles) {
    __shared__ float sU[KCL * DDIM * DDIM];   // 16 KB
    __shared__ float sNB[KCL * DDIM];         // 1 KB  (-b)
    __shared__ float sC[KCL];

    for (int i = threadIdx.x; i < KCL * DDIM * DDIM; i += 256) sU[i] = Ug[i];
    if (threadIdx.x < KCL * DDIM) sNB[threadIdx.x] = nbg[threadIdx.x];
    if (threadIdx.x < KCL)        sC[threadIdx.x] = cg[threadIdx.x];
    __syncthreads();

    const int wave = threadIdx.x >> 5;
    const int lane = threadIdx.x & 31;
    const int n = lane & 15;     // point within tile / matrix column
    const int h = lane >> 4;     // lane half: selects K-pair & row half

    // hoist cluster constants into registers (once per wave; only 16 regs)
    float ck[KCL];
#pragma unroll
    for (int k = 0; k < KCL; ++k) ck[k] = sC[k];

    const int stride = gridDim.x * 8;
    for (int tile = blockIdx.x * 8 + wave; tile < ntiles; tile += stride) {
        // Opaque zero, redefined every iteration: blocks LICM of LDS reads.
        int zoff;
        asm volatile("s_mov_b32 %0, 0" : "=s"(zoff));

        // B operand (X^T tile): lane holds B[K=4kk+2h+{0,1}][n] = x[p][4kk+2h+{0,1}]
        // streaming data: non-temporal (read once)
        const v2f* xp = (const v2f*)(x + (size_t)(tile * 16 + n) * DDIM + 2 * h);
        v2f bx[4];
#pragma unroll
        for (int kk = 0; kk < 4; ++kk)
            bx[kk] = __builtin_nontemporal_load(xp + kk * 2);

        float logit[KCL];

#pragma unroll
        for (int k = 0; k < KCL; ++k) {
            // C/D layout: VGPR v, half h -> row M = v + 8h.  Seed with -b_k[M].
            const float* bp = sNB + zoff + k * DDIM + 8 * h;
            v8f acc;
#pragma unroll
            for (int v = 0; v < 8; ++v) acc[v] = bp[v];

            // A operand (U_k): lane holds A[M=n][K=4kk+2h+{0,1}]
            const float* up = sU + zoff + k * DDIM * DDIM + n * DDIM + 2 * h;
#pragma unroll
            for (int kk = 0; kk < 4; ++kk) {
                v2f a = *(const v2f*)(up + kk * 4);
                acc = __builtin_amdgcn_wmma_f32_16x16x4_f32(
                    false, a, false, bx[kk], (short)0, acc, false, false);
            }

            float s = 0.0f;
#pragma unroll
            for (int v = 0; v < 8; ++v) s += acc[v] * acc[v];
            s += __shfl_xor(s, 16, 32);          // add the other 8 rows (wave32)
            logit[k] = ck[k] - 0.5f * s;
        }

        // softmax over K in registers (values duplicated across lane halves)
        float mx = logit[0];
#pragma unroll
        for (int k = 1; k < KCL; ++k) mx = fmaxf(mx, logit[k]);
        float e[KCL], sum = 0.0f;
#pragma unroll
        for (int k = 0; k < KCL; ++k) { e[k] = expf(logit[k] - mx); sum += e[k]; }
        float inv = 1.0f / sum;

        // half h writes k = 8h .. 8h+7 of its point's output row
        float o[8];
#pragma unroll
        for (int j = 0; j < 8; ++j) o[j] = (h ? e[8 + j] : e[j]) * inv;

        v4f* op = (v4f*)(out + (size_t)(tile * 16 + n) * KCL + 8 * h);
        __builtin_nontemporal_store((v4f){o[0], o[1], o[2], o[3]}, op);
        __builtin_nontemporal_store((v4f){o[4], o[5], o[6], o[7]}, op + 1);
    }
}

// ---------------------------------------------------------------------------
extern "C" void kernel_launch(void* const* d_in, const int* in_sizes, int n_in,
                              void* d_out, int out_size, void* d_ws, size_t ws_size,
                              hipStream_t stream) {
    const float* x     = (const float*)d_in[0];   // (N,16)
    const float* pi    = (const float*)d_in[1];   // (1,16)
    const float* mu    = (const float*)d_in[2];   // (16,1,16)
    const float* Sigma = (const float*)d_in[3];   // (16,16,16)
    float* out = (float*)d_out;                   // (N,16)

    float* Uw  = (float*)d_ws;                    // 4096 floats
    float* nbw = Uw + KCL * DDIM * DDIM;          // 256 floats (-b)
    float* cw  = nbw + KCL * DDIM;                // 16 floats

    gmm_prep_kernel<<<1, 32, 0, stream>>>(pi, mu, Sigma, Uw, nbw, cw);

    const int ntiles = NTILES;
    const int blocks = 1024;                      // 8192 waves, ~4 tiles each
    gmm_main_kernel<<<blocks, 256, 0, stream>>>(x, Uw, nbw, cw, out, ntiles);
}